// NeuralDecisionTree_74603581931538
// MI455X (gfx1250) — compile-verified
//
#include <hip/hip_runtime.h>

typedef float v2f __attribute__((ext_vector_type(2)));
typedef float v8f __attribute__((ext_vector_type(8)));

#define FCH 32

// ---------------- exact sort-based entmax15 over rows (for feature masks) ----
__global__ __launch_bounds__(512)
void ndt_entmax_rows(const float* __restrict__ fm, float* __restrict__ mask, int Fd)
{
    __shared__ float sv[512];
    __shared__ float tauArr[320];
    __shared__ float s_tau, s_max;
    const int row = blockIdx.x;
    const int t = threadIdx.x;

    float xval = (t < Fd) ? fm[(size_t)row * Fd + t] * 0.5f : -1.0e30f;
    sv[t] = xval;
    __syncthreads();

    // bitonic sort, descending
    for (int k = 2; k <= 512; k <<= 1) {
        for (int j = k >> 1; j > 0; j >>= 1) {
            int ixj = t ^ j;
            if (ixj > t) {
                float a = sv[t], b = sv[ixj];
                bool doswap = ((t & k) == 0) ? (a < b) : (a > b);
                if (doswap) { sv[t] = b; sv[ixj] = a; }
            }
            __syncthreads();
        }
    }

    if (t == 0) {
        float mx = sv[0];
        float S = 0.0f, S2 = 0.0f;
        int cnt = 0;
        for (int k = 0; k < Fd; ++k) {
            float xs = sv[k] - mx;
            S += xs; S2 += xs * xs;
            float rho = (float)(k + 1);
            float mean = S / rho;
            float msq = S2 / rho;
            float ss = rho * (msq - mean * mean);
            float tau = mean - sqrtf(fmaxf(0.0f, (1.0f - ss) / rho));
            tauArr[k] = tau;
            if (tau <= xs) cnt++;
        }
        s_tau = tauArr[cnt - 1];
        s_max = mx;
    }
    __syncthreads();

    if (t < Fd) {
        float v = fmaxf(xval - s_max - s_tau, 0.0f);
        mask[(size_t)row * Fd + t] = v * v;
    }
}

// closed-form 2-element entmax15 for logits (x-cp0, 2x-cp1) after internal /2.
// Clamping delta to 1 reproduces the support==1 branch exactly (at=1 -> (1,0)).
__device__ __forceinline__ v2f entmax2(float xv, float q)
{
    float tt = xv - q;                                 // t>=0 -> l=1 wins
    float at = fminf(fabsf(tt) * 0.5f, 1.0f);          // clamped delta
    float s  = __builtin_amdgcn_sqrtf(__builtin_fmaf(at, -0.25f * at, 0.5f));
    float hi = __builtin_fmaf(0.5f, at, s);  hi *= hi; // p of winning bin
    float lo = __builtin_fmaf(-0.5f, at, s); lo *= lo; // p of losing bin
    v2f p;
    p.x = (tt >= 0.0f) ? lo : hi;                      // l = 0
    p.y = (tt >= 0.0f) ? hi : lo;                      // l = 1
    return p;
}

// ---------------- per-level stump kernel: closed-form 2-entmax + WMMA reduce -
// PREV = columns from previous level's output (0,2,4,8,16,32)
// WAVES = waves per block; each wave owns a 16-row b-tile.
template<int PREV, int WAVES>
__global__ __launch_bounds__(WAVES * 32)
void ndt_stump_wmma(const float* __restrict__ x,
                    const float* __restrict__ prev,
                    const float* __restrict__ cp, const float* __restrict__ lr,
                    const float* __restrict__ mask,
                    float* __restrict__ out, int outCols, int Fd)
{
    constexpr int THREADS = WAVES * 32;
    constexpr int ROWS = WAVES * 16;
    constexpr int LD = (ROWS * (FCH / 4)) / THREADS;   // float4 loads per thread (=4)
    __shared__ float xt[2][ROWS][FCH + 4];             // double buffer, 16B-aligned rows
    __shared__ __align__(16) float qv[320];
    __shared__ __align__(16) float c0v[320];
    __shared__ __align__(16) float c1v[320];

    const int n    = blockIdx.x;
    const int b0   = blockIdx.y * ROWS;
    const int t    = threadIdx.x;
    const int lane = t & 31;
    const int wave = t >> 5;
    const int half = lane >> 4;                 // 0: f0 (K0/K1), 1: f1 (K2/K3)
    const int brow = (wave << 4) + (lane & 15); // b-row within block tile

    // stage stump parameters: q = cp1-cp0, c_l = lr_l * mask
    const size_t pbase = (size_t)n * Fd;
    for (int f = t; f < Fd; f += THREADS) {
        float m  = mask[pbase + f];
        float a0 = cp[(pbase + f) * 2 + 0];
        float a1 = cp[(pbase + f) * 2 + 1];
        qv[f]  = a1 - a0;
        c0v[f] = lr[(pbase + f) * 2 + 0] * m;
        c1v[f] = lr[(pbase + f) * 2 + 1] * m;
    }

    // uniform B-operand build: lane 0 feeds column 0 (c0), lane 17 column 1 (c1)
    const float bmask = (lane == 0 || lane == 17) ? 1.0f : 0.0f;
    const float* cbase = (lane == 17) ? c1v : c0v;

    v8f acc = {0.f, 0.f, 0.f, 0.f, 0.f, 0.f, 0.f, 0.f};
    float4 stage[LD];

    auto loadChunk = [&](int fc) {
        #pragma unroll
        for (int i = 0; i < LD; ++i) {
            int k = t + i * THREADS;
            int r = k >> 3, f4 = k & 7;        // 8 float4 per 32-feature row chunk
            stage[i] = reinterpret_cast<const float4*>(
                           x + (size_t)(b0 + r) * 256 + fc)[f4];
        }
    };
    auto storeChunk = [&](int buf) {
        #pragma unroll
        for (int i = 0; i < LD; ++i) {
            int k = t + i * THREADS;
            int r = k >> 3, f4 = k & 7;
            reinterpret_cast<float4*>(&xt[buf][r][0])[f4] = stage[i];
        }
    };

    // ---- phase 1: original x features [0,256), double-buffered 32-f chunks --
    loadChunk(0);
    int buf = 0;
    for (int fc = 0; fc < 256; fc += FCH) {
        storeChunk(buf);
        __syncthreads();                        // also covers param staging on fc==0
        if (fc + FCH < 256) loadChunk(fc + FCH);
        #pragma unroll
        for (int fi = 0; fi < FCH; fi += 2) {
            int fl = fi + half;
            int fg = fc + fi;
            v2f A = entmax2(xt[buf][brow][fl], qv[fc + fl]);
            float2 cc = reinterpret_cast<const float2*>(cbase)[fg >> 1];
            v2f Bv; Bv.x = cc.x * bmask; Bv.y = cc.y * bmask;
            acc = __builtin_amdgcn_wmma_f32_16x16x4_f32(false, A, false, Bv,
                                                        (short)0, acc, false, false);
        }
        buf ^= 1;
    }

    // ---- phase 2: previous-level outputs, features [256, 256+PREV) ----------
    if constexpr (PREV > 0) {
        __syncthreads();                        // finish all phase-1 reads
        #pragma unroll
        for (int k = t; k < ROWS * PREV; k += THREADS) {
            int r = k / PREV, ff = k % PREV;    // power-of-two constants
            xt[0][r][ff] = prev[(size_t)(b0 + r) * PREV + ff];
        }
        __syncthreads();
        #pragma unroll
        for (int fi = 0; fi < PREV; fi += 2) {
            int fl = fi + half;
            int fg = 256 + fi;
            v2f A = entmax2(xt[0][brow][fl], qv[256 + fl]);
            float2 cc = reinterpret_cast<const float2*>(cbase)[fg >> 1];
            v2f Bv; Bv.x = cc.x * bmask; Bv.y = cc.y * bmask;
            acc = __builtin_amdgcn_wmma_f32_16x16x4_f32(false, A, false, Bv,
                                                        (short)0, acc, false, false);
        }
    }

    // ---- store: C columns 0 (l=0) and 1 (l=1); VGPR r -> M=r (+8 high half) -
    int col = lane & 15;
    if (col < 2) {
        int rbase = (wave << 4) + (half << 3);
        float* p = out + (size_t)(b0 + rbase) * outCols + (n * 2 + col);
        #pragma unroll
        for (int r = 0; r < 8; ++r) { *p = acc[r]; p += outCols; }
    }
}

// ---------------- launch -----------------------------------------------------
extern "C" void kernel_launch(void* const* d_in, const int* in_sizes, int n_in,
                              void* d_out, int out_size, void* d_ws, size_t ws_size,
                              hipStream_t stream)
{
    (void)in_sizes; (void)n_in; (void)out_size; (void)ws_size;
    const int B = 8192, F = 256, DEPTH = 6;

    const float* x = (const float*)d_in[0];
    float* out = (float*)d_out;
    float* ws  = (float*)d_ws;

    int ns[DEPTH], Fds[DEPTH];
    for (int d = 0; d < DEPTH; ++d) {
        ns[d]  = 1 << d;
        Fds[d] = F + ((d > 0) ? (1 << d) : 0);
    }

    // d_out layout: [ out (B x 64) | mask0 | ... | mask5 ]
    size_t mOff[DEPTH];
    size_t off = (size_t)B * 64;
    for (int d = 0; d < DEPTH; ++d) { mOff[d] = off; off += (size_t)ns[d] * Fds[d]; }

    // workspace: per-level intermediate outputs for levels 0..4
    size_t wOff[DEPTH - 1];
    size_t woff = 0;
    for (int d = 0; d < DEPTH - 1; ++d) { wOff[d] = woff; woff += (size_t)B * (2 << d); }

    const float* prev = nullptr;
    for (int d = 0; d < DEPTH; ++d) {
        const float* fm = (const float*)d_in[1 + 3 * d];
        const float* cp = (const float*)d_in[2 + 3 * d];
        const float* lr = (const float*)d_in[3 + 3 * d];
        float* maskp = out + mOff[d];

        ndt_entmax_rows<<<ns[d], 512, 0, stream>>>(fm, maskp, Fds[d]);

        float* lvlOut = (d == DEPTH - 1) ? out : (ws + wOff[d]);
        int outCols = 2 * ns[d];
        int Fd = Fds[d];

        switch (d) {
        case 0: {   // shallow levels: 2-wave blocks spread across more WGPs
            dim3 g(1, B / 32);
            ndt_stump_wmma<0, 2><<<g, 64, 0, stream>>>(x, prev, cp, lr, maskp, lvlOut, outCols, Fd);
        } break;
        case 1: {
            dim3 g(2, B / 32);
            ndt_stump_wmma<2, 2><<<g, 64, 0, stream>>>(x, prev, cp, lr, maskp, lvlOut, outCols, Fd);
        } break;
        case 2: {
            dim3 g(4, B / 32);
            ndt_stump_wmma<4, 2><<<g, 64, 0, stream>>>(x, prev, cp, lr, maskp, lvlOut, outCols, Fd);
        } break;
        case 3: {
            dim3 g(8, B / 128);
            ndt_stump_wmma<8, 8><<<g, 256, 0, stream>>>(x, prev, cp, lr, maskp, lvlOut, outCols, Fd);
        } break;
        case 4: {
            dim3 g(16, B / 128);
            ndt_stump_wmma<16, 8><<<g, 256, 0, stream>>>(x, prev, cp, lr, maskp, lvlOut, outCols, Fd);
        } break;
        default: {
            dim3 g(32, B / 128);
            ndt_stump_wmma<32, 8><<<g, 256, 0, stream>>>(x, prev, cp, lr, maskp, lvlOut, outCols, Fd);
        } break;
        }
        prev = lvlOut;
    }
}